// CDIRecModel_13005160972813
// MI455X (gfx1250) — compile-verified
//
#include <hip/hip_runtime.h>
#include <math.h>
#include <stdint.h>

typedef __attribute__((ext_vector_type(2))) float v2f;
typedef __attribute__((ext_vector_type(8))) float v8f;

#define D_  5
#define B_  2048
#define L_  50
#define DH_ 128
#define DU_ 128
#define QK_ 64
#define CD_ 64
#define INV_SCALE 0.125f   /* 1/sqrt(64) */
#define EPS_ 1e-5f

__device__ __forceinline__ void atomAddF(float* p, float v) {
    __hip_atomic_fetch_add(p, v, __ATOMIC_RELAXED, __HIP_MEMORY_SCOPE_AGENT);
}

// gfx1250 async global->LDS copy (ASYNCcnt-tracked, bypasses VGPRs).
// lds_off: byte offset within the workgroup's LDS allocation (per-lane VGPR)
// gptr   : per-lane 64-bit global address
__device__ __forceinline__ void async_copy_b128(uint32_t lds_off, const void* gptr) {
    asm volatile("global_load_async_to_lds_b128 %0, %1, off"
                 :: "v"(lds_off), "v"(gptr) : "memory");
}
__device__ __forceinline__ void wait_async0() {
    asm volatile("s_wait_asynccnt 0x0" ::: "memory");
}
__device__ __forceinline__ uint32_t lds_offset_of(const void* shared_generic) {
    // generic LDS addresses carry the LDS byte offset in the low 32 bits
    return (uint32_t)(uintptr_t)shared_generic;
}

// ---------------------------------------------------------------------------
// K1: one workgroup per batch element b. Async-stages H tiles into LDS once,
// attention via the GEMV factorization (Q.K = (Wk.Q).H, h = (a^T H).Wv), then
// one fp32 WMMA GEMM for z = [h_i|h_s] @ Wf. Contrastive partials + segment
// sums via native f32 global atomics.
// ---------------------------------------------------------------------------
__launch_bounds__(256)
__global__ void k1_attention(
    const float* __restrict__ u,  const float* __restrict__ H_intra,
    const float* __restrict__ Hs, const float* __restrict__ h,
    const int*   __restrict__ dom,
    const float* __restrict__ Wq,  const float* __restrict__ bq,
    const float* __restrict__ Wki, const float* __restrict__ bki,
    const float* __restrict__ Wvi, const float* __restrict__ bvi,
    const float* __restrict__ Wks, const float* __restrict__ bks,
    const float* __restrict__ Wvs, const float* __restrict__ bvs,
    const float* __restrict__ Wf,  const float* __restrict__ bf,
    float* __restrict__ zsel, float* __restrict__ Lacc,
    float* __restrict__ cnt,  float* __restrict__ sumh, float* __restrict__ sumh2)
{
    __shared__ float sH [L_ * DH_];       // current domain tile (25.6 KB)
    __shared__ float sHS[L_ * DH_];       // shared-history tile (25.6 KB)
    __shared__ float sHC[16][2 * QK_];    // padded [h_i|h_s] A-matrix (8 KB)
    __shared__ float sU [DU_];
    __shared__ float sC [DH_];
    __shared__ float sQ [QK_];
    __shared__ float sKt[DH_];
    __shared__ float sSc[L_];
    __shared__ float sAH[DH_];
    __shared__ float sQB[2];
    __shared__ float sZ  [D_][CD_];       // z rows (bf included)
    __shared__ float sAll[D_][D_];
    __shared__ float sPos[D_];

    const int b    = blockIdx.x;
    const int tid  = threadIdx.x;
    const int lane = tid & 31;
    const int wave = tid >> 5;

    for (int i = tid; i < DU_; i += 256) sU[i] = u[b * DU_ + i];
    // zero pad rows 5..15 of the WMMA A-matrix once
    for (int i = tid; i < (16 - D_) * 2 * QK_; i += 256)
        (&sHC[D_][0])[i] = 0.0f;
    {   // async-stage H_share[b] (1600 x B128)
        const char* g = (const char*)(Hs + (size_t)b * L_ * DH_);
        const uint32_t l0 = lds_offset_of(&sHS[0]);
        for (int i = tid; i < L_ * DH_ / 4; i += 256)
            async_copy_b128(l0 + i * 16, g + (size_t)i * 16);
    }

    for (int d = 0; d < D_; ++d) {
        {   // async-stage H_intra[d,b]
            const char* g = (const char*)(H_intra + ((size_t)(d * B_ + b)) * L_ * DH_);
            const uint32_t l0 = lds_offset_of(&sH[0]);
            for (int i = tid; i < L_ * DH_ / 4; i += 256)
                async_copy_b128(l0 + i * 16, g + (size_t)i * 16);
        }
        wait_async0();
        __syncthreads();
        // c = mean over L
        if (tid < DH_) {
            float s = 0.f;
            for (int l = 0; l < L_; ++l) s += sH[l * DH_ + tid];
            sC[tid] = s * (1.0f / L_);
        }
        __syncthreads();
        // Q = [u;c] @ Wq + bq
        if (tid < QK_) {
            float s = bq[tid];
            for (int k = 0; k < DU_; ++k) s += sU[k] * Wq[k * QK_ + tid];
            for (int k = 0; k < DH_; ++k) s += sC[k] * Wq[(DU_ + k) * QK_ + tid];
            sQ[tid] = s;
        }
        __syncthreads();
        if (tid == 0) {
            float s0 = 0.f, s1 = 0.f;
            for (int q = 0; q < QK_; ++q) { s0 += sQ[q] * bki[q]; s1 += sQ[q] * bks[q]; }
            sQB[0] = s0; sQB[1] = s1;
        }
        // k~ = Wki @ Q
        if (tid < DH_) {
            float s = 0.f;
            for (int q = 0; q < QK_; ++q) s += Wki[tid * QK_ + q] * sQ[q];
            sKt[tid] = s;
        }
        __syncthreads();
        // intra scores
        if (tid < L_) {
            float s = sQB[0];
            for (int j = 0; j < DH_; ++j) s += sH[tid * DH_ + j] * sKt[j];
            sSc[tid] = s * INV_SCALE;
        }
        __syncthreads();
        if (tid == 0) {
            float m = -1e30f;
            for (int l = 0; l < L_; ++l) m = fmaxf(m, sSc[l]);
            float sum = 0.f;
            for (int l = 0; l < L_; ++l) { float e = __expf(sSc[l] - m); sSc[l] = e; sum += e; }
            float inv = 1.0f / sum;
            for (int l = 0; l < L_; ++l) sSc[l] *= inv;
        }
        __syncthreads();
        if (tid < DH_) {
            float s = 0.f;
            for (int l = 0; l < L_; ++l) s += sSc[l] * sH[l * DH_ + tid];
            sAH[tid] = s;
        }
        __syncthreads();
        // h_i = aH @ Wvi + bvi  -> A-matrix row d, cols [0,64)
        if (tid < QK_) {
            float s = bvi[tid];
            for (int j = 0; j < DH_; ++j) s += sAH[j] * Wvi[j * QK_ + tid];
            sHC[d][tid] = s;
        }
        __syncthreads();
        // ---- shared-history attention (same Q) ----
        if (tid < DH_) {
            float s = 0.f;
            for (int q = 0; q < QK_; ++q) s += Wks[tid * QK_ + q] * sQ[q];
            sKt[tid] = s;
        }
        __syncthreads();
        if (tid < L_) {
            float s = sQB[1];
            for (int j = 0; j < DH_; ++j) s += sHS[tid * DH_ + j] * sKt[j];
            sSc[tid] = s * INV_SCALE;
        }
        __syncthreads();
        if (tid == 0) {
            float m = -1e30f;
            for (int l = 0; l < L_; ++l) m = fmaxf(m, sSc[l]);
            float sum = 0.f;
            for (int l = 0; l < L_; ++l) { float e = __expf(sSc[l] - m); sSc[l] = e; sum += e; }
            float inv = 1.0f / sum;
            for (int l = 0; l < L_; ++l) sSc[l] *= inv;
        }
        __syncthreads();
        if (tid < DH_) {
            float s = 0.f;
            for (int l = 0; l < L_; ++l) s += sSc[l] * sHS[l * DH_ + tid];
            sAH[tid] = s;
        }
        __syncthreads();
        // h_s -> A-matrix row d, cols [64,128)
        if (tid < QK_) {
            float s = bvs[tid];
            for (int j = 0; j < DH_; ++j) s += sAH[j] * Wvs[j * QK_ + tid];
            sHC[d][QK_ + tid] = s;
        }
        __syncthreads();
    }

    // ---- z = sHC (16x128) @ Wf (128x64) via fp32 WMMA, 4 waves x 1 N-tile ----
    if (wave < 4) {
        const int m    = lane & 15;          // A row / N col within tile
        const int half = lane >> 4;          // K-pair selector (ISA layout)
        const int ncol = wave * 16 + m;
        v8f acc = {};
        for (int k0 = 0; k0 < 2 * QK_; k0 += 4) {
            const int ka = k0 + half * 2;    // even -> 8B aligned
            v2f a = *(const v2f*)&sHC[m][ka];
            v2f bb;
            bb.x = Wf[ka * CD_ + ncol];
            bb.y = Wf[(ka + 1) * CD_ + ncol];
            acc = __builtin_amdgcn_wmma_f32_16x16x4_f32(false, a, false, bb,
                                                        (short)0, acc, false, false);
        }
        if (half == 0) {   // lanes 0-15 hold rows 0..7 in acc[0..7]; need rows 0..4
            const float bfv = bf[ncol];
            sZ[0][ncol] = acc[0] + bfv;
            sZ[1][ncol] = acc[1] + bfv;
            sZ[2][ncol] = acc[2] + bfv;
            sZ[3][ncol] = acc[3] + bfv;
            sZ[4][ncol] = acc[4] + bfv;
        }
    }
    __syncthreads();

    // ---- contrastive partials (normalize in place AFTER z used raw values) ----
    if (tid < 2 * D_) {
        float* v = (tid < D_) ? &sHC[tid][0] : &sHC[tid - D_][QK_];
        float s = 0.f;
        for (int q = 0; q < QK_; ++q) s += v[q] * v[q];
        float inv = 1.0f / fmaxf(sqrtf(s), 1e-12f);
        for (int q = 0; q < QK_; ++q) v[q] *= inv;
    }
    __syncthreads();
    if (tid < D_ * D_) {
        const int di = tid / D_, e = tid % D_;
        float s = 0.f;
        for (int q = 0; q < QK_; ++q) s += sHC[di][q] * sHC[e][q];
        sAll[di][e] = s;
    }
    if (tid >= 32 && tid < 32 + D_) {
        const int di = tid - 32;
        float s = 0.f;
        for (int q = 0; q < QK_; ++q) s += sHC[di][q] * sHC[di][QK_ + q];
        sPos[di] = s;
    }
    __syncthreads();
    if (tid == 0) {
        float c = 0.f;
        for (int di = 0; di < D_; ++di) {
            float den = 0.f;
            for (int e = 0; e < D_; ++e) den += __expf(sAll[di][e]);
            c += logf(__expf(sPos[di]) / (den + 1e-8f) + 1e-8f);
        }
        atomAddF(Lacc, c);
    }

    // ---- z_sel + domain segment sums ----
    const int ds = dom[b];
    if (tid < CD_) zsel[b * CD_ + tid] = sZ[ds][tid];
    if (tid < DH_) {
        const float hv = h[b * DH_ + tid];
        atomAddF(&sumh [ds * DH_ + tid], hv);
        atomAddF(&sumh2[ds * DH_ + tid], hv * hv);
    }
    if (tid == 0) atomAddF(&cnt[ds], 1.0f);
}

// ---------------------------------------------------------------------------
// K2: 16 batch rows per workgroup. gb = z_sel @ Wgb via fp32 WMMA, then the
// domain-masked AdaNorm using segment sums from K1.
// ---------------------------------------------------------------------------
__launch_bounds__(256)
__global__ void k2_inject(
    const float* __restrict__ h, const int* __restrict__ dom,
    const float* __restrict__ Wgb, const float* __restrict__ bgb,
    const float* __restrict__ zsel,
    const float* __restrict__ cnt, const float* __restrict__ sumh,
    const float* __restrict__ sumh2, float* __restrict__ out)
{
    __shared__ float sZ[16][QK_];
    __shared__ float sGB[16][2 * DH_];

    const int tid  = threadIdx.x;
    const int lane = tid & 31;
    const int wave = tid >> 5;
    const int b0   = blockIdx.x * 16;

    for (int i = tid; i < 16 * QK_; i += 256)
        sZ[i >> 6][i & 63] = zsel[b0 * QK_ + i];
    __syncthreads();

    // 16 N-tiles of 16 cols over 8 waves -> 2 tiles per wave; K=64 -> 16 steps
    const int m    = lane & 15;
    const int half = lane >> 4;
    for (int t = 0; t < 2; ++t) {
        const int ncol = (wave * 2 + t) * 16 + m;
        v8f acc = {};
        for (int k0 = 0; k0 < QK_; k0 += 4) {
            const int ka = k0 + half * 2;    // even -> 8B aligned
            v2f a = *(const v2f*)&sZ[m][ka];
            v2f bb;
            bb.x = Wgb[ka * (2 * DH_) + ncol];
            bb.y = Wgb[(ka + 1) * (2 * DH_) + ncol];
            acc = __builtin_amdgcn_wmma_f32_16x16x4_f32(false, a, false, bb,
                                                        (short)0, acc, false, false);
        }
#pragma unroll
        for (int r = 0; r < 8; ++r) {
            const int row = half * 8 + r;     // D layout: lanes16-31 hold rows 8..15
            sGB[row][ncol] = acc[r];
        }
    }
    __syncthreads();

    for (int i = tid; i < 16 * DH_; i += 256) {
        const int row = i >> 7;
        const int j   = i & 127;
        const int b   = b0 + row;
        const int ds  = dom[b];
        const float c  = fmaxf(cnt[ds], 1.0f);
        const float mu = sumh[ds * DH_ + j] / c;
        const float va = sumh2[ds * DH_ + j] / c - mu * mu;
        const float hv = h[b * DH_ + j];
        const float hn = (hv - mu) * rsqrtf(va + EPS_);
        const float gamma = sGB[row][j] + bgb[j];
        const float beta  = sGB[row][DH_ + j] + bgb[DH_ + j];
        out[b * DH_ + j] = gamma * hn + beta + hv;
    }
}

__global__ void k3_finalize(const float* __restrict__ Lacc, float* __restrict__ out_scalar)
{
    if (threadIdx.x == 0 && blockIdx.x == 0)
        out_scalar[0] = -Lacc[0] * (1.0f / (float)(B_ * D_));
}

// ---------------------------------------------------------------------------
extern "C" void kernel_launch(void* const* d_in, const int* in_sizes, int n_in,
                              void* d_out, int out_size, void* d_ws, size_t ws_size,
                              hipStream_t stream)
{
    (void)in_sizes; (void)n_in; (void)out_size; (void)ws_size;
    const float* u       = (const float*)d_in[0];
    const float* H_intra = (const float*)d_in[1];
    const float* H_share = (const float*)d_in[2];
    const float* h       = (const float*)d_in[3];
    const int*   dom     = (const int*)  d_in[4];
    const float* Wq  = (const float*)d_in[5];  const float* bq  = (const float*)d_in[6];
    const float* Wki = (const float*)d_in[7];  const float* bki = (const float*)d_in[8];
    const float* Wvi = (const float*)d_in[9];  const float* bvi = (const float*)d_in[10];
    const float* Wks = (const float*)d_in[11]; const float* bks = (const float*)d_in[12];
    const float* Wvs = (const float*)d_in[13]; const float* bvs = (const float*)d_in[14];
    const float* Wf  = (const float*)d_in[15]; const float* bf  = (const float*)d_in[16];
    const float* Wgb = (const float*)d_in[17]; const float* bgb = (const float*)d_in[18];

    float* ws    = (float*)d_ws;
    float* Lacc  = ws;                 // 1   (padded)
    float* cnt   = ws + 8;             // 5 -> pad to 8
    float* sumh  = ws + 16;            // 640
    float* sumh2 = ws + 16 + 640;      // 640
    float* zsel  = ws + 16 + 1280;     // B*CD = 131072
    float* outv  = (float*)d_out;      // [B*DH] then scalar L_con

    hipMemsetAsync(d_ws, 0, (16 + 1280) * sizeof(float), stream);

    k1_attention<<<B_, 256, 0, stream>>>(u, H_intra, H_share, h, dom,
                                         Wq, bq, Wki, bki, Wvi, bvi,
                                         Wks, bks, Wvs, bvs, Wf, bf,
                                         zsel, Lacc, cnt, sumh, sumh2);
    k2_inject<<<B_ / 16, 256, 0, stream>>>(h, dom, Wgb, bgb, zsel,
                                           cnt, sumh, sumh2, outv);
    k3_finalize<<<1, 1, 0, stream>>>(Lacc, outv + (size_t)B_ * DH_);
}